// QuantumLayer_57466662420516
// MI455X (gfx1250) — compile-verified
//
#include <hip/hip_runtime.h>
#include <math.h>

typedef __attribute__((ext_vector_type(2))) float v2f;
typedef __attribute__((ext_vector_type(8))) float v8f;

#define NQ   12
#define NL   4
#define DIM  4096          // 2^12 amplitudes
#define TPB  128           // 4 wave32s
#define WAVES 4
#define PI_F 3.14159265358979323846f

// ---------------------------------------------------------------------------
// Kernel 1: build the 12 (= 4 layers x 3 groups) fused 16x16 complex gate
// matrices from the 144 variational angles.  G = C_q0 (x) C_q1 (x) C_q2 (x) C_q3
// with C = RZ * RY * RX (RX applied first).  Output to workspace:
//   gre[(l*3+g)*256 + r*16 + c], gim[...] (row r = output index, col c = input)
// ---------------------------------------------------------------------------
__global__ void build_gates_kernel(const float* __restrict__ qw,
                                   float* __restrict__ gre,
                                   float* __restrict__ gim) {
    __shared__ float Cre[48][4];   // [l*12+q][r*2+c]
    __shared__ float Cim[48][4];
    int t = threadIdx.x;
    if (t < 48) {
        int l = t / 12, q = t % 12;
        const float* p = qw + l * 36 + q * 3;
        float tx = 0.5f * p[0], ty = 0.5f * p[1], tz = 0.5f * p[2];
        float cx = cosf(tx), sx = sinf(tx);
        float cy = cosf(ty), sy = sinf(ty);
        float cz = cosf(tz), sz = sinf(tz);
        // M = RY * RX   (RX = [[cx, -i sx],[-i sx, cx]], RY = [[cy,-sy],[sy,cy]])
        float mr[4], mi[4];
        mr[0] =  cy * cx;  mi[0] =  sy * sx;
        mr[1] = -sy * cx;  mi[1] = -cy * sx;
        mr[2] =  sy * cx;  mi[2] = -cy * sx;
        mr[3] =  cy * cx;  mi[3] = -sy * sx;
        // C = RZ * M : row0 *= (cz - i sz), row1 *= (cz + i sz)
        for (int c = 0; c < 2; ++c) {
            float ar = mr[c],     ai = mi[c];
            Cre[t][c]     = cz * ar + sz * ai;
            Cim[t][c]     = cz * ai - sz * ar;
            float br = mr[2 + c], bi = mi[2 + c];
            Cre[t][2 + c] = cz * br - sz * bi;
            Cim[t][2 + c] = cz * bi + sz * br;
        }
    }
    __syncthreads();
    for (int e = t; e < 12 * 256; e += blockDim.x) {
        int mat = e >> 8;            // l*3 + g
        int l   = mat / 3, g = mat % 3;
        int rc  = e & 255;
        int r   = rc >> 4, c = rc & 15;
        float vr = 1.0f, vi = 0.0f;
        #pragma unroll
        for (int j = 0; j < 4; ++j) {
            int q  = 4 * g + j;
            int rb = (r >> (3 - j)) & 1;
            int cb = (c >> (3 - j)) & 1;
            int id = l * 12 + q;
            float er = Cre[id][rb * 2 + cb];
            float ei = Cim[id][rb * 2 + cb];
            float nr = vr * er - vi * ei;
            float ni = vr * ei + vi * er;
            vr = nr; vi = ni;
        }
        gre[e] = vr;
        gim[e] = vi;
    }
}

// Flat amplitude index for group G (compile-time) with 4-bit group index k and
// 8-bit rest n -> pure constant shifts, no branches.
template <int G>
__device__ __forceinline__ int gaddr(int k, int n) {
    if constexpr (G == 0) {
        return (k << 8) | n;
    } else if constexpr (G == 1) {
        return ((n & 0xF0) << 4) | (k << 4) | (n & 15);
    } else {
        return (n << 4) | k;
    }
}

// Apply the fused 16x16 complex gate of group G to the LDS state.
// D = G x B (complex) via 4 real f32 WMMA chains (K=4 x 4 slices).
template <int G>
__device__ __forceinline__ void apply_group(const float* __restrict__ Gr,
                                            const float* __restrict__ Gi,
                                            float* __restrict__ sre,
                                            float* __restrict__ sim_,
                                            int wave, int lhi, int llo) {
    // A fragments: element v of slice s holds G[M=llo][K = 4s + v + 2*lhi]
    v2f Ar[4], Ai[4], Ani[4];
    #pragma unroll
    for (int s = 0; s < 4; ++s) {
        int k0 = 4 * s + 2 * lhi;
        Ar[s].x  =  Gr[llo * 16 + k0];
        Ar[s].y  =  Gr[llo * 16 + k0 + 1];
        Ai[s].x  =  Gi[llo * 16 + k0];
        Ai[s].y  =  Gi[llo * 16 + k0 + 1];
        Ani[s].x = -Ai[s].x;
        Ani[s].y = -Ai[s].y;
    }
    // 16 column tiles of the 16x256 state view; 4 consecutive tiles per wave.
    // unroll-2: enough ILP to pipeline LDS->WMMA, stays well under 256 VGPRs
    // so 8 blocks/WGP (LDS-limited) remain resident.
    #pragma unroll 2
    for (int tile = 0; tile < 4; ++tile) {
        int n = (wave * 4 + tile) * 16 + llo;
        v2f Br[4], Bi[4];
        #pragma unroll
        for (int s = 0; s < 4; ++s) {
            int k0 = 4 * s + 2 * lhi;
            int a0 = gaddr<G>(k0,     n);
            int a1 = gaddr<G>(k0 + 1, n);
            Br[s].x = sre[a0];  Br[s].y = sre[a1];
            Bi[s].x = sim_[a0]; Bi[s].y = sim_[a1];
        }
        v8f Dre = {};
        v8f Dim = {};
        #pragma unroll
        for (int s = 0; s < 4; ++s)
            Dre = __builtin_amdgcn_wmma_f32_16x16x4_f32(
                false, Ar[s], false, Br[s], (short)0, Dre, false, false);
        #pragma unroll
        for (int s = 0; s < 4; ++s)
            Dre = __builtin_amdgcn_wmma_f32_16x16x4_f32(
                false, Ani[s], false, Bi[s], (short)0, Dre, false, false);
        #pragma unroll
        for (int s = 0; s < 4; ++s)
            Dim = __builtin_amdgcn_wmma_f32_16x16x4_f32(
                false, Ar[s], false, Bi[s], (short)0, Dim, false, false);
        #pragma unroll
        for (int s = 0; s < 4; ++s)
            Dim = __builtin_amdgcn_wmma_f32_16x16x4_f32(
                false, Ai[s], false, Br[s], (short)0, Dim, false, false);
        // D element v: row M = v + 8*lhi, col N = llo
        #pragma unroll
        for (int v = 0; v < 8; ++v) {
            int a = gaddr<G>(v + 8 * lhi, n);
            sre[a]  = Dre[v];
            sim_[a] = Dim[v];
        }
    }
}

// ---------------------------------------------------------------------------
// Kernel 2: one workgroup per sample.  State (re/im) lives in LDS.
// ---------------------------------------------------------------------------
__global__ __launch_bounds__(TPB)
void qsim_kernel(const float* __restrict__ x,
                 const float* __restrict__ gre,
                 const float* __restrict__ gim,
                 const float* __restrict__ dw,
                 const float* __restrict__ db,
                 float* __restrict__ out) {
    __shared__ float sre[DIM];
    __shared__ float sim_[DIM];
    __shared__ float cs[24];            // cos/sin of half encoding angles
    __shared__ float aux[TPB * 12];     // per-thread <Z> partials
    __shared__ float qv[12];

    const int samp = blockIdx.x;
    const int tid  = threadIdx.x;
    const int wave = tid >> 5;
    const int lane = tid & 31;
    const int lhi  = lane >> 4;         // 0/1 : lane half
    const int llo  = lane & 15;

    // ---- encoding angles (l2-normalize, angle scale), serial but tiny ----
    if (tid == 0) {
        float xv[NQ], ss = 0.0f;
        for (int w = 0; w < NQ; ++w) { xv[w] = x[samp * NQ + w]; ss += xv[w] * xv[w]; }
        float inv = rsqrtf(fmaxf(ss, 1e-12f));
        float mx = 0.0f;
        for (int w = 0; w < NQ; ++w) { xv[w] *= inv; mx = fmaxf(mx, fabsf(xv[w])); }
        float sc = PI_F / (mx + 1e-8f);
        for (int w = 0; w < NQ; ++w) {
            float a = 0.5f * sc * xv[w];
            cs[w]      = cosf(a);
            cs[12 + w] = sinf(a);
        }
    }
    __syncthreads();

    // ---- encoding on |0..0> is a product state: psi[i] = prod(bit? s : c) ----
    // i = tid*32 + j : wires 0..6 (bit pos 11..5) depend only on tid.
    {
        float cw[NQ], sw[NQ];
        #pragma unroll
        for (int w = 0; w < NQ; ++w) { cw[w] = cs[w]; sw[w] = cs[12 + w]; }
        float base = 1.0f;
        #pragma unroll
        for (int w = 0; w < 7; ++w)
            base *= ((tid >> (6 - w)) & 1) ? sw[w] : cw[w];
        float4* re4 = (float4*)(sre  + tid * 32);
        float4* im4 = (float4*)(sim_ + tid * 32);
        #pragma unroll
        for (int j4 = 0; j4 < 8; ++j4) {
            float e[4];
            #pragma unroll
            for (int u = 0; u < 4; ++u) {
                int j = j4 * 4 + u;
                float v = base;
                #pragma unroll
                for (int w = 7; w < 12; ++w)
                    v *= ((j >> (11 - w)) & 1) ? sw[w] : cw[w];
                e[u] = v;
            }
            re4[j4] = make_float4(e[0], e[1], e[2], e[3]);
            im4[j4] = make_float4(0.f, 0.f, 0.f, 0.f);
        }
    }
    __syncthreads();

    // ---- variational layers ----
    for (int l = 0; l < NL; ++l) {
        const float* Grb = gre + l * 3 * 256;
        const float* Gib = gim + l * 3 * 256;
        apply_group<0>(Grb,       Gib,       sre, sim_, wave, lhi, llo);
        __syncthreads();
        apply_group<1>(Grb + 256, Gib + 256, sre, sim_, wave, lhi, llo);
        __syncthreads();
        apply_group<2>(Grb + 512, Gib + 512, sre, sim_, wave, lhi, llo);
        __syncthreads();

        // ---- CNOT ring: closed-form composed permutation (GF(2)-linear) ----
        //   src bits 0..9 : x_p ^ x_{p+1}
        //   src bit  10   : x10 ^ x11 ^ x0
        //   src bit  11   : x11 ^ x0
        //   => src = i ^ (i>>1) ^ ((i&1) ? 0xC00 : 0)
        float tre[32], tim[32];
        #pragma unroll 4
        for (int j = 0; j < 32; ++j) {
            int i   = tid * 32 + j;
            int src = (i ^ (i >> 1)) ^ ((-(i & 1)) & 0xC00);
            tre[j] = sre[src];
            tim[j] = sim_[src];
        }
        __syncthreads();
        {
            float4* re4 = (float4*)(sre  + tid * 32);
            float4* im4 = (float4*)(sim_ + tid * 32);
            #pragma unroll
            for (int j4 = 0; j4 < 8; ++j4) {
                re4[j4] = make_float4(tre[4 * j4], tre[4 * j4 + 1],
                                      tre[4 * j4 + 2], tre[4 * j4 + 3]);
                im4[j4] = make_float4(tim[4 * j4], tim[4 * j4 + 1],
                                      tim[4 * j4 + 2], tim[4 * j4 + 3]);
            }
        }
        __syncthreads();
    }

    // ---- measurement: <Z_w> = sum_i |psi_i|^2 * (1 - 2*bit_w) ----
    // wires 0..6: sign fixed per thread -> one signed copy of the total.
    // wires 7..11: sign depends on j only -> 5 conditional accumulators.
    {
        float Ptot = 0.0f;
        float accL[5] = {0.f, 0.f, 0.f, 0.f, 0.f};
        const float4* re4 = (const float4*)(sre  + tid * 32);
        const float4* im4 = (const float4*)(sim_ + tid * 32);
        #pragma unroll
        for (int j4 = 0; j4 < 8; ++j4) {
            float4 r = re4[j4];
            float4 m = im4[j4];
            float pr[4] = { r.x * r.x + m.x * m.x, r.y * r.y + m.y * m.y,
                            r.z * r.z + m.z * m.z, r.w * r.w + m.w * m.w };
            #pragma unroll
            for (int u = 0; u < 4; ++u) {
                int j = j4 * 4 + u;
                float p = pr[u];
                Ptot += p;
                #pragma unroll
                for (int w = 7; w < 12; ++w)
                    accL[w - 7] += ((j >> (11 - w)) & 1) ? -p : p;
            }
        }
        #pragma unroll
        for (int w = 0; w < 7; ++w)
            aux[tid * 12 + w] = ((tid >> (6 - w)) & 1) ? -Ptot : Ptot;
        #pragma unroll
        for (int w = 7; w < 12; ++w)
            aux[tid * 12 + w] = accL[w - 7];
    }
    __syncthreads();
    if (tid < 12) {
        float s = 0.0f;
        for (int t2 = 0; t2 < TPB; ++t2) s += aux[t2 * 12 + tid];
        qv[tid] = s;
    }
    __syncthreads();
    // ---- dense head + tanh ----
    if (tid < 12) {
        float z = db[tid];
        #pragma unroll
        for (int w = 0; w < 12; ++w) z += qv[w] * dw[w * 12 + tid];
        out[samp * 12 + tid] = tanhf(z);
    }
}

// ---------------------------------------------------------------------------
extern "C" void kernel_launch(void* const* d_in, const int* in_sizes, int n_in,
                              void* d_out, int out_size, void* d_ws, size_t ws_size,
                              hipStream_t stream) {
    const float* x  = (const float*)d_in[0];   // (4096, 12)
    const float* qw = (const float*)d_in[1];   // (144,)
    const float* dw = (const float*)d_in[2];   // (12, 12)
    const float* db = (const float*)d_in[3];   // (12,)
    float* out = (float*)d_out;                // (4096, 12)

    float* gre = (float*)d_ws;                 // 12 * 256 floats
    float* gim = gre + 12 * 256;               // 12 * 256 floats (24 KB total)

    build_gates_kernel<<<1, 128, 0, stream>>>(qw, gre, gim);
    qsim_kernel<<<4096, TPB, 0, stream>>>(x, gre, gim, dw, db, out);

    (void)in_sizes; (void)n_in; (void)out_size; (void)ws_size;
}